// HT_CUDA_21431886807082
// MI455X (gfx1250) — compile-verified
//
#include <hip/hip_runtime.h>
#include <stdint.h>

// ---------------- problem geometry ----------------
#define IM_P      16384          // 128*128 pixels per map
#define HT_B      33120          // 184*180 Hough bins
#define BC_MAPS   32             // 2*16 independent maps
#define NORM_INV  (1.0f / 128.0f)

// ---------------- kernel tiling -------------------
#define BLOCK     256            // 8 wave32s
#define TILE      1024           // votes staged per tile
#define VPT       (TILE / BLOCK) // 4 votes per thread per tile
#define CHUNKS    8              // vote-chunks per map (grid.x)

// ---------------- LDS layout (dynamic) ------------
// [ x map 64KB ][ accumulator 132480B ][ stage buf0 16KB ][ stage buf1 16KB ]
// Stage buffers hold interleaved 16B vote slots {pix, bin, w, pad} so the
// consumer needs a single ds_load_b128 per vote.
#define SH_X_BYTES    (IM_P * 4)                 // 65536
#define SH_ACC_BYTES  (HT_B * 4)                 // 132480
#define STG_BYTES     (TILE * 16)                // 16384 per buffer
#define LDS_BYTES     (SH_X_BYTES + SH_ACC_BYTES + 2 * STG_BYTES)  // 230784 < 320KB

// CDNA5 async global->LDS copies (ASYNCcnt-tracked, per-lane addresses).
// Generic pointers to __shared__ carry the LDS byte offset in their low 32
// bits, so truncation yields the LDS-address VGPR the instruction needs.
__device__ __forceinline__ void async_copy_b32(void* lds_dst, const void* gsrc) {
  asm volatile("global_load_async_to_lds_b32 %0, %1, off"
               :: "v"((unsigned int)(unsigned long long)lds_dst), "v"(gsrc)
               : "memory");
}
__device__ __forceinline__ void async_copy_b128(void* lds_dst, const void* gsrc) {
  asm volatile("global_load_async_to_lds_b128 %0, %1, off"
               :: "v"((unsigned int)(unsigned long long)lds_dst), "v"(gsrc)
               : "memory");
}
#define WAIT_ASYNCCNT(imm) asm volatile("s_wait_asynccnt " #imm ::: "memory")

__global__ void __launch_bounds__(BLOCK)
ht_zero_out(float* __restrict__ out, int n) {
  int i = blockIdx.x * BLOCK + threadIdx.x;
  if (i < n) out[i] = 0.0f;
}

__global__ void __launch_bounds__(BLOCK)
ht_vote_kernel(const float* __restrict__ x,
               const int*   __restrict__ vpix,
               const int*   __restrict__ vbin,
               const float* __restrict__ vw,
               float*       __restrict__ out,
               int n_votes) {
  const int tid = threadIdx.x;
  const int bc  = blockIdx.y;          // which of the 32 maps
  const int chk = blockIdx.x;          // which vote chunk

  extern __shared__ char smem[];
  float* sh_x   = (float*)smem;                                // 16384 f32
  float* sh_acc = (float*)(smem + SH_X_BYTES);                 // 33120 f32
  char*  stg    = smem + SH_X_BYTES + SH_ACC_BYTES;            // 2 stage bufs

  // ---- stage this map's x (64KB) into LDS via async DMA (b128 granularity)
  const float* xmap = x + (size_t)bc * IM_P;
  for (int i = tid; i < IM_P / 4; i += BLOCK)
    async_copy_b128(&sh_x[4 * i], xmap + 4 * i);
  for (int i = tid; i < HT_B; i += BLOCK)
    sh_acc[i] = 0.0f;
  WAIT_ASYNCCNT(0);
  __syncthreads();

  // ---- this block's vote range
  const int per = (n_votes + CHUNKS - 1) / CHUNKS;     // 500000
  const int v0  = chk * per;
  const int v1  = (v0 + per < n_votes) ? (v0 + per) : n_votes;
  const int ntile = (v1 - v0 + TILE - 1) / TILE;

  const int last = n_votes - 1;

  // Stage tile t into buffer (t&1): exactly 12 async-load instructions per
  // wave (3 fields x 4 votes/thread), addresses clamped so every wave always
  // issues the full count -> exact ASYNCcnt bookkeeping, tail-independent.
  // The three SoA fields are scattered into one interleaved 16B LDS slot.
#define STAGE(t)                                                            \
  do {                                                                      \
    char* sbase = stg + ((t) & 1) * STG_BYTES;                              \
    int base = v0 + (t) * TILE;                                             \
    _Pragma("unroll")                                                       \
    for (int k = 0; k < VPT; ++k) {                                         \
      int s   = tid + k * BLOCK;                                            \
      int idx = base + s;                                                   \
      idx = (idx <= last) ? idx : last;                                     \
      char* slot = sbase + (size_t)s * 16;                                  \
      async_copy_b32(slot + 0, vpix + idx);                                 \
      async_copy_b32(slot + 4, vbin + idx);                                 \
      async_copy_b32(slot + 8, vw + idx);                                   \
    }                                                                       \
  } while (0)

  if (ntile > 0) STAGE(0);

  for (int t = 0; t < ntile; ++t) {
    // double buffer: keep next tile in flight, retire oldest 12 async loads
    if (t + 1 < ntile) {
      STAGE(t + 1);
      WAIT_ASYNCCNT(12);
    } else {
      WAIT_ASYNCCNT(0);
    }
    // barrier-free consume: each thread reads exactly the votes it staged
    char* sbase = stg + (t & 1) * STG_BYTES;
    int base = v0 + t * TILE;
#pragma unroll
    for (int k = 0; k < VPT; ++k) {
      int s = tid + k * BLOCK;
      int4 vt = *(const int4*)(sbase + (size_t)s * 16);   // one ds_load_b128
      if (base + s < v1) {
        int   p = vt.x;
        int   b = vt.y;
        float w = __int_as_float(vt.z);
        atomicAdd(&sh_acc[b], sh_x[p] * w);   // ds_add_f32, LDS-privatized
      }
    }
  }
#undef STAGE

  __syncthreads();
  // ---- flush privatized histogram; fold the /NORM into the flush
  float* omap = out + (size_t)bc * HT_B;
  for (int i = tid; i < HT_B; i += BLOCK)
    atomicAdd(&omap[i], sh_acc[i] * NORM_INV);
}

extern "C" void kernel_launch(void* const* d_in, const int* in_sizes, int n_in,
                              void* d_out, int out_size, void* d_ws, size_t ws_size,
                              hipStream_t stream) {
  const float* x    = (const float*)d_in[0];   // [2,16,128,128] f32
  const int*   vpix = (const int*)d_in[1];     // [N] int32
  const int*   vbin = (const int*)d_in[2];     // [N] int32
  const float* vw   = (const float*)d_in[3];   // [N] f32
  float*       out  = (float*)d_out;           // [2,16,184,180] f32
  const int n_votes = in_sizes[1];

  // harness poisons d_out; flush uses atomics, so zero it first
  ht_zero_out<<<(out_size + BLOCK - 1) / BLOCK, BLOCK, 0, stream>>>(out, out_size);

  // allow >64KB dynamic LDS (deterministic, not a stream op)
  hipFuncSetAttribute((const void*)ht_vote_kernel,
                      hipFuncAttributeMaxDynamicSharedMemorySize, LDS_BYTES);

  dim3 grid(CHUNKS, BC_MAPS);   // 8 vote-chunks x 32 maps = 256 workgroups
  ht_vote_kernel<<<grid, BLOCK, LDS_BYTES, stream>>>(x, vpix, vbin, vw, out, n_votes);
}